// RRWPPositionalEncoding_61160334295119
// MI455X (gfx1250) — compile-verified
//
#include <hip/hip_runtime.h>

#define KPOW 16

typedef __attribute__((ext_vector_type(2))) float v2f;
typedef __attribute__((ext_vector_type(8))) float v8f;

// ---------------------------------------------------------------- zero
__global__ void k_zero(float* __restrict__ p, int n) {
  int i = blockIdx.x * blockDim.x + threadIdx.x;
  int stride = gridDim.x * blockDim.x;
  for (; i < n; i += stride) p[i] = 0.0f;
}

// ------------------------------------------------- scatter-add edges into A
__global__ void k_scatter(float* __restrict__ A, const float* __restrict__ w,
                          const long long* __restrict__ src,
                          const long long* __restrict__ dst, int E, int N) {
  int e = blockIdx.x * blockDim.x + threadIdx.x;
  if (e >= E) return;
  int s = (int)src[e];
  int d = (int)dst[e];
  __hip_atomic_fetch_add(&A[(size_t)s * N + d], w[e],
                         __ATOMIC_RELAXED, __HIP_MEMORY_SCOPE_AGENT);
}

// ------------------------- row-normalize with fused self-loop: M = (A+I)/deg
__global__ __launch_bounds__(256) void k_rownorm(const float* __restrict__ A,
                                                 float* __restrict__ M, int N) {
  int row = blockIdx.x;
  int tid = threadIdx.x;
  const float* a = A + (size_t)row * N;
  float* m = M + (size_t)row * N;

  float vals[8];
  float s = 0.0f;
#pragma unroll
  for (int t = 0; t < 8; ++t) {
    int j = tid + t * 256;
    float v = 0.0f;
    if (j < N) v = a[j] + ((j == row) ? 1.0f : 0.0f);
    vals[t] = v;
    s += v;
  }

  __shared__ float red[256];
  red[tid] = s;
  __syncthreads();
  for (int off = 128; off > 0; off >>= 1) {
    if (tid < off) red[tid] += red[tid + off];
    __syncthreads();
  }
  float inv = 1.0f / red[0];  // deg >= 1 because of the self loop

#pragma unroll
  for (int t = 0; t < 8; ++t) {
    int j = tid + t * 256;
    if (j < N) m[j] = vals[t] * inv;
  }
}

// ------------------------------------------- powers 0 (=I) and 1 (=M) -> out
__global__ void k_init01(const float* __restrict__ M,
                         const long long* __restrict__ src,
                         const long long* __restrict__ dst,
                         float* __restrict__ out, int N, int E) {
  int i = blockIdx.x * blockDim.x + threadIdx.x;
  if (i < N) {
    out[(size_t)i * KPOW + 0] = 1.0f;
    out[(size_t)i * KPOW + 1] = M[(size_t)i * N + i];
  }
  int e = i - N;
  if (e >= 0 && e < E) {
    int s = (int)src[e];
    int d = (int)dst[e];
    float* ep = out + (size_t)N * KPOW + (size_t)e * KPOW;
    ep[0] = (s == d) ? 1.0f : 0.0f;
    ep[1] = M[(size_t)s * N + d];
  }
}

// --------------------------------------- per-power gather of diag + edges
__global__ void k_gather(const float* __restrict__ Tk,
                         const long long* __restrict__ src,
                         const long long* __restrict__ dst,
                         float* __restrict__ out, int N, int E, int k) {
  int i = blockIdx.x * blockDim.x + threadIdx.x;
  if (i < N) out[(size_t)i * KPOW + k] = Tk[(size_t)i * N + i];
  int e = i - N;
  if (e >= 0 && e < E) {
    int s = (int)src[e];
    int d = (int)dst[e];
    out[(size_t)N * KPOW + (size_t)e * KPOW + k] = Tk[(size_t)s * N + d];
  }
}

// ------------------------------------------------------------------ GEMM
// C = T * M, all 2048x2048 f32 row-major, via V_WMMA_F32_16X16X4_F32.
// Block tile 64x128, 8 waves; each wave -> 16 rows x 64 cols (4 v8f accs).
// A tile filled with GLOBAL_LOAD_ASYNC_TO_LDS_B128 (ASYNCcnt path);
// B tile staged TRANSPOSED so each B fragment is one aligned b64 LDS read.
#define BM 64
#define BN 128
#define BK 32
#define AST 36  // As row stride in floats (pad, keeps 16B alignment)
#define BTS 34  // Bt column-record stride in floats (even -> b64 aligned)

__global__ __launch_bounds__(256) void k_gemm(const float* __restrict__ T,
                                              const float* __restrict__ Mx,
                                              float* __restrict__ C, int N) {
  __shared__ float As[BM * AST];  // 9216 B, row-major rows x K
  __shared__ float Bt[BN * BTS];  // 17408 B, TRANSPOSED: col-record x K

  const int tid  = threadIdx.x;
  const int lane = tid & 31;
  const int wave = tid >> 5;
  const int wm = wave & 3;   // 4 waves along M: rows wm*16
  const int wn = wave >> 2;  // 2 waves along N: cols wn*64
  const int bm = blockIdx.y * BM;
  const int bn = blockIdx.x * BN;

  const int lrow = lane & 15;  // A: matrix row / B,C,D: column
  const int lhi  = lane >> 4;  // 0|1 -> K offset (A/B), row offset +8 (C/D)

  v8f acc[4];
  {
    v8f z = {};
#pragma unroll
    for (int s = 0; s < 4; ++s) acc[s] = z;
  }

  for (int k0 = 0; k0 < N; k0 += BK) {
    // ---- A tile 64x32: async copy global -> LDS, 128b per lane ----
#pragma unroll
    for (int i = 0; i < 2; ++i) {
      int idx = tid + i * 256;  // 0..511
      int r = idx >> 3;         // row 0..63
      int c = (idx & 7) << 2;   // col 0..28 step 4
      // low 32 bits of the flat shared-aperture address == LDS byte offset
      unsigned lds = (unsigned)(uintptr_t)&As[r * AST + c];
      const float* g = &T[(size_t)(bm + r) * N + k0 + c];
      asm volatile("global_load_async_to_lds_b128 %0, %1, off"
                   :: "v"(lds), "v"(g)
                   : "memory");
    }
    // ---- B tile 32x128: load row-major, store transposed ----
#pragma unroll
    for (int i = 0; i < 4; ++i) {
      int idx = tid + i * 256;  // 0..1023
      int r = idx >> 5;         // k-row 0..31
      int c = (idx & 31) << 2;  // col 0..124 step 4
      float4 v = *(const float4*)&Mx[(size_t)(k0 + r) * N + bn + c];
      Bt[(c + 0) * BTS + r] = v.x;
      Bt[(c + 1) * BTS + r] = v.y;
      Bt[(c + 2) * BTS + r] = v.z;
      Bt[(c + 3) * BTS + r] = v.w;
    }
#if __has_builtin(__builtin_amdgcn_s_wait_asynccnt)
    __builtin_amdgcn_s_wait_asynccnt(0);
#else
    asm volatile("s_wait_asynccnt 0" ::: "memory");
#endif
    __syncthreads();

#pragma unroll
    for (int kk = 0; kk < BK; kk += 4) {
      // A fragment 16x4: lane holds T[bm+wm*16+lrow][kk + 2*lhi + {0,1}]
      v2f a = *(const v2f*)&As[(wm * 16 + lrow) * AST + kk + (lhi << 1)];
#pragma unroll
      for (int s = 0; s < 4; ++s) {
        int colb = wn * 64 + s * 16 + lrow;
        // B fragment 4x16: one aligned b64 read from transposed tile
        v2f b = *(const v2f*)&Bt[colb * BTS + kk + (lhi << 1)];
        acc[s] = __builtin_amdgcn_wmma_f32_16x16x4_f32(
            false, a, false, b, (short)0, acc[s], false, false);
      }
    }
    __syncthreads();
  }

  // store C: VGPR v, lanes 0-15 -> row v, lanes 16-31 -> row v+8
#pragma unroll
  for (int s = 0; s < 4; ++s) {
    int col = bn + wn * 64 + s * 16 + lrow;
#pragma unroll
    for (int v = 0; v < 8; ++v) {
      int row = bm + wm * 16 + v + (lhi << 3);
      C[(size_t)row * N + col] = acc[s][v];
    }
  }
}

// ------------------------------------------------------------------ driver
extern "C" void kernel_launch(void* const* d_in, const int* in_sizes, int n_in,
                              void* d_out, int out_size, void* d_ws,
                              size_t ws_size, hipStream_t stream) {
  const float* w = (const float*)d_in[0];
  const long long* ei = (const long long*)d_in[1];  // int64 [2, E]
  const int E = in_sizes[0];
  const int N = out_size / KPOW - E;  // 2048 for the reference shapes
  const long long* src = ei;
  const long long* dst = ei + E;
  float* out = (float*)d_out;

  char* ws = (char*)d_ws;
  const size_t matBytes = (size_t)N * N * sizeof(float);
  float* P0 = (float*)(ws);                 // adjacency, later ping buffer
  float* Mx = (float*)(ws + matBytes);      // normalized transition matrix
  float* P2 = (float*)(ws + 2 * matBytes);  // pong buffer

  // 1. zero adjacency
  k_zero<<<1024, 256, 0, stream>>>(P0, N * N);
  // 2. scatter edge weights
  k_scatter<<<(E + 255) / 256, 256, 0, stream>>>(P0, w, src, dst, E, N);
  // 3. add self loops + row-normalize -> Mx
  k_rownorm<<<N, 256, 0, stream>>>(P0, Mx, N);
  // 4. powers 0 and 1
  int tot = N + E;
  k_init01<<<(tot + 255) / 256, 256, 0, stream>>>(Mx, src, dst, out, N, E);
  // 5. serial GEMM chain: T_k = T_{k-1} * M, gather diag + edge entries
  dim3 grid(N / BN, N / BM), blk(256);
  const float* T = Mx;
  float* ping[2] = {P2, P0};
  for (int k = 2; k < KPOW; ++k) {
    float* Cn = ping[k & 1];
    k_gemm<<<grid, blk, 0, stream>>>(T, Mx, Cn, N);
    k_gather<<<(tot + 255) / 256, 256, 0, stream>>>(Cn, src, dst, out, N, E, k);
    T = Cn;
  }
}